// GATModel_46170898432490
// MI455X (gfx1250) — compile-verified
//
#include <hip/hip_runtime.h>
#include <hip/hip_bf16.h>
#include <math.h>

typedef __attribute__((ext_vector_type(2))) float v2f;
typedef __attribute__((ext_vector_type(8))) float v8f;

#define NEG_SLOPE 0.2f
#define NINF (-__builtin_inff())

// ---------- atomics (native f32 max/add via clang HIP atomic builtins) ----------
__device__ __forceinline__ void atomicMaxF(float* p, float v) {
    __hip_atomic_fetch_max(p, v, __ATOMIC_RELAXED, __HIP_MEMORY_SCOPE_AGENT);
}
__device__ __forceinline__ void atomicAddF(float* p, float v) {
    __hip_atomic_fetch_add(p, v, __ATOMIC_RELAXED, __HIP_MEMORY_SCOPE_AGENT);
}

// ---------- pad W [K x 41] (row-major) into Bp [Kpad x 48], zeros elsewhere ----------
__global__ void gat_padB(const float* __restrict__ W, int K, int ldb, int Ncols,
                         float* __restrict__ Bp, int Kpad) {
    int idx = blockIdx.x * blockDim.x + threadIdx.x;
    if (idx >= Kpad * 48) return;
    int k = idx / 48, c = idx - k * 48;
    Bp[idx] = (k < K && c < Ncols) ? W[(size_t)k * ldb + c] : 0.f;
}

// ---------- WMMA fp32 GEMM: Hout[M x 48] = A[M x K] (stride lda) * Bp[Kpad x 48] ----------
// One wave computes a 16x48 output tile (3 WMMA accumulators share the A fragment).
// M must be a multiple of 16 (100000 = 6250*16). Uses V_WMMA_F32_16X16X4_F32.
__global__ __launch_bounds__(128)
void gat_gemm_wmma(const float* __restrict__ A, int lda, int K,
                   const float* __restrict__ Bp,
                   float* __restrict__ Hout, int M) {
    const int wave  = threadIdx.x >> 5;
    const int lane  = threadIdx.x & 31;
    const int mTile = blockIdx.x * 4 + wave;
    if (mTile * 16 >= M) return;          // wave-uniform: EXEC stays all-1s

    const int r16 = lane & 15;            // A row-in-tile / B,C column-in-tile
    const int khi = lane >> 4;            // 0: K+0/+1, 1: K+2/+3 (A & B layouts)
    const float* Arow = A + (size_t)(mTile * 16 + r16) * lda;
    const float* Bcol = Bp + r16;

    v8f c0 = {}, c1 = {}, c2 = {};
    int k0 = 0;
    for (; k0 + 4 <= K; k0 += 4) {        // guard-free hot loop (Bp is pre-padded)
        const int ka = k0 + 2 * khi;
        v2f a; a.x = Arow[ka]; a.y = Arow[ka + 1];
        const float* Bk = Bcol + (size_t)ka * 48;
        v2f b0; b0.x = Bk[0];  b0.y = Bk[48];
        v2f b1; b1.x = Bk[16]; b1.y = Bk[64];
        v2f b2; b2.x = Bk[32]; b2.y = Bk[80];
        c0 = __builtin_amdgcn_wmma_f32_16x16x4_f32(false, a, false, b0, (short)0, c0, false, false);
        c1 = __builtin_amdgcn_wmma_f32_16x16x4_f32(false, a, false, b1, (short)0, c1, false, false);
        c2 = __builtin_amdgcn_wmma_f32_16x16x4_f32(false, a, false, b2, (short)0, c2, false, false);
    }
    if (k0 < K) {                         // K % 4 remainder: only A needs guards
        const int ka = k0 + 2 * khi;
        v2f a;
        a.x = (ka     < K) ? Arow[ka]     : 0.f;
        a.y = (ka + 1 < K) ? Arow[ka + 1] : 0.f;
        const float* Bk = Bcol + (size_t)ka * 48;
        v2f b0; b0.x = Bk[0];  b0.y = Bk[48];
        v2f b1; b1.x = Bk[16]; b1.y = Bk[64];
        v2f b2; b2.x = Bk[32]; b2.y = Bk[80];
        c0 = __builtin_amdgcn_wmma_f32_16x16x4_f32(false, a, false, b0, (short)0, c0, false, false);
        c1 = __builtin_amdgcn_wmma_f32_16x16x4_f32(false, a, false, b1, (short)0, c1, false, false);
        c2 = __builtin_amdgcn_wmma_f32_16x16x4_f32(false, a, false, b2, (short)0, c2, false, false);
    }
    // C/D layout: VGPR r -> row (r + 8*khi), column = r16 (+ tile offset)
    float* O = Hout + (size_t)(mTile * 16 + 8 * khi) * 48 + r16;
#pragma unroll
    for (int r = 0; r < 8; ++r) {
        O[(size_t)r * 48 +  0] = c0[r];
        O[(size_t)r * 48 + 16] = c1[r];
        O[(size_t)r * 48 + 32] = c2[r];
    }
}

// ---------- per-layer scratch init ----------
__global__ void gat_init(float* __restrict__ emax, float* __restrict__ denom,
                         float* __restrict__ aggr, int n48) {
    int i = blockIdx.x * blockDim.x + threadIdx.x;
    if (i >= n48) return;
    emax[i] = NINF; denom[i] = 0.f; aggr[i] = NINF;
}

// edge logit for (edge e, head hh): leaky_relu(h[s]*a_s + h[d]*a_d)   (C == 1)
__device__ __forceinline__ float edge_logit(const float* __restrict__ h,
                                            const float* __restrict__ a_s,
                                            const float* __restrict__ a_d,
                                            int s, int d, int hh) {
    float v = h[(size_t)s * 48 + hh] * a_s[hh] + h[(size_t)d * 48 + hh] * a_d[hh];
    return (v > 0.f) ? v : NEG_SLOPE * v;
}

// ---------- edge pass 1: segment max of logits over dst ----------
__global__ __launch_bounds__(480)
void gat_edge_max(const int* __restrict__ srcA, const int* __restrict__ dstA,
                  int E, int Etot, const float* __restrict__ h,
                  const float* __restrict__ a_s, const float* __restrict__ a_d,
                  float* __restrict__ emax) {
    int t = blockIdx.x * 480 + threadIdx.x;
    int e = t / 48, hh = t - e * 48;
    if (e >= Etot || hh >= 41) return;
    int s, d;
    if (e < E) { s = srcA[e]; d = dstA[e]; } else { s = e - E; d = s; }
    float v = edge_logit(h, a_s, a_d, s, d, hh);
    atomicMaxF(&emax[(size_t)d * 48 + hh], v);
}

// ---------- edge pass 2: segment sum of exp(logit - max) ----------
__global__ __launch_bounds__(480)
void gat_edge_sum(const int* __restrict__ srcA, const int* __restrict__ dstA,
                  int E, int Etot, const float* __restrict__ h,
                  const float* __restrict__ a_s, const float* __restrict__ a_d,
                  const float* __restrict__ emax, float* __restrict__ denom) {
    int t = blockIdx.x * 480 + threadIdx.x;
    int e = t / 48, hh = t - e * 48;
    if (e >= Etot || hh >= 41) return;
    int s, d;
    if (e < E) { s = srcA[e]; d = dstA[e]; } else { s = e - E; d = s; }
    float v  = edge_logit(h, a_s, a_d, s, d, hh);
    float ex = __expf(v - emax[(size_t)d * 48 + hh]);
    atomicAddF(&denom[(size_t)d * 48 + hh], ex);
}

// ---------- edge pass 3: max-aggregate alpha * h[src] over dst ----------
__global__ __launch_bounds__(480)
void gat_edge_aggr(const int* __restrict__ srcA, const int* __restrict__ dstA,
                   int E, int Etot, const float* __restrict__ h,
                   const float* __restrict__ a_s, const float* __restrict__ a_d,
                   const float* __restrict__ emax, const float* __restrict__ denom,
                   float* __restrict__ aggr) {
    int t = blockIdx.x * 480 + threadIdx.x;
    int e = t / 48, hh = t - e * 48;
    if (e >= Etot || hh >= 41) return;
    int s, d;
    if (e < E) { s = srcA[e]; d = dstA[e]; } else { s = e - E; d = s; }
    size_t di = (size_t)d * 48 + hh;
    float v     = edge_logit(h, a_s, a_d, s, d, hh);
    float alpha = __expf(v - emax[di]) / denom[di];
    float m     = alpha * h[(size_t)s * 48 + hh];
    atomicMaxF(&aggr[di], m);
}

// ---------- finalize: + bias, optional relu, write padded next-layer input ----------
__global__ __launch_bounds__(480)
void gat_finalize(const float* __restrict__ aggr, const float* __restrict__ bias,
                  float* __restrict__ xnext, int relu, int N) {
    int idx = blockIdx.x * 480 + threadIdx.x;
    int n = idx / 48, hh = idx - n * 48;
    if (n >= N) return;
    float v = 0.f;
    if (hh < 41) {
        v = aggr[idx] + bias[hh];
        if (relu) v = fmaxf(v, 0.f);
    }
    xnext[idx] = v;
}

// ---------- log_softmax over the 41 classes ----------
__global__ void gat_logsoftmax(const float* __restrict__ x, float* __restrict__ out, int N) {
    int n = blockIdx.x * blockDim.x + threadIdx.x;
    if (n >= N) return;
    const float* r = x + (size_t)n * 48;
    float mx = NINF;
    for (int j = 0; j < 41; ++j) mx = fmaxf(mx, r[j]);
    float ssum = 0.f;
    for (int j = 0; j < 41; ++j) ssum += expf(r[j] - mx);
    float lse = mx + logf(ssum);
    float* o = out + (size_t)n * 41;
    for (int j = 0; j < 41; ++j) o[j] = r[j] - lse;
}

extern "C" void kernel_launch(void* const* d_in, const int* in_sizes, int n_in,
                              void* d_out, int out_size, void* d_ws, size_t ws_size,
                              hipStream_t stream) {
    const float* x     = (const float*)d_in[0];
    const int*   ei    = (const int*)  d_in[1];
    const float* W0    = (const float*)d_in[2];
    const float* Wrest = (const float*)d_in[3];
    const float* attS  = (const float*)d_in[4];
    const float* attD  = (const float*)d_in[5];
    const float* bias  = (const float*)d_in[6];
    float* out = (float*)d_out;

    const int N = 100000, F_IN = 602, L = 5;
    const int E = in_sizes[1] / 2;          // 1,600,000
    const int Etot = E + N;                 // + self loops
    const int* srcA = ei;
    const int* dstA = ei + E;

    const int n48 = N * 48;
    float* p = (float*)d_ws;
    float* hbuf  = p; p += n48;             // per-head features  [N x 48]
    float* emax  = p; p += n48;             // segment max logits
    float* denom = p; p += n48;             // segment sum exp
    float* aggr  = p; p += n48;             // max-aggregated messages
    float* xbuf  = p; p += n48;             // next-layer input (padded)
    float* Bp    = p;                       // padded weight panel [604 x 48]

    const float* xin = x;
    int lda = F_IN;
    const int mBlocks = (N / 16 + 3) / 4;   // 4 waves (m-tiles) per block
    const int eBlocks = (Etot + 9) / 10;    // 10 edges x 48 threads per block

    for (int l = 0; l < L; ++l) {
        const float* W = (l == 0) ? W0 : (Wrest + (size_t)(l - 1) * 41 * 41);
        const int K = (l == 0) ? F_IN : 41;
        const int Kpad = (K + 3) & ~3;

        gat_padB<<<(Kpad * 48 + 255) / 256, 256, 0, stream>>>(W, K, 41, 41, Bp, Kpad);
        gat_gemm_wmma<<<mBlocks, 128, 0, stream>>>(xin, lda, K, Bp, hbuf, N);
        gat_init<<<(n48 + 255) / 256, 256, 0, stream>>>(emax, denom, aggr, n48);

        const float* as = attS + l * 41;
        const float* ad = attD + l * 41;
        gat_edge_max <<<eBlocks, 480, 0, stream>>>(srcA, dstA, E, Etot, hbuf, as, ad, emax);
        gat_edge_sum <<<eBlocks, 480, 0, stream>>>(srcA, dstA, E, Etot, hbuf, as, ad, emax, denom);
        gat_edge_aggr<<<eBlocks, 480, 0, stream>>>(srcA, dstA, E, Etot, hbuf, as, ad, emax, denom, aggr);

        gat_finalize<<<(n48 + 479) / 480, 480, 0, stream>>>(aggr, bias + l * 41, xbuf,
                                                            (l != L - 1) ? 1 : 0, N);
        xin = xbuf; lda = 48;
    }
    gat_logsoftmax<<<(N + 255) / 256, 256, 0, stream>>>(xbuf, out, N);
}